// IndRNN_onlyrecurrent_80333068304562
// MI455X (gfx1250) — compile-verified
//
#include <hip/hip_runtime.h>

// IndRNN recurrence: h_t = relu(h_{t-1} * w + x_t), elementwise over (B,H),
// sequential over T. Pure streaming: 268 MB read + 268 MB write -> ~23 us
// floor at 23.3 TB/s. Strategy: 48-deep CDNA5 async global->LDS prefetch
// pipeline per wave (ASYNCcnt), 4x-unrolled consume to keep issue cost well
// under the memory floor, NT stores for the write-once output stream.

#define TT      1024
#define BB      64
#define HH      1024
#define NN      (BB * HH)      // 65536 independent recurrence lanes
#define THREADS 256
#define DEPTH   48             // outstanding async copies per wave (ASYNCcnt max 63)
#define RING    64             // LDS ring slots (pow2 > DEPTH -> safe reuse distance)
#define UNROLL  4              // consumes per wait; (TT-DEPTH) % UNROLL == 0

#if __has_builtin(__builtin_amdgcn_global_load_async_to_lds_b32) && \
    __has_builtin(__builtin_amdgcn_s_wait_asynccnt)
#define USE_ASYNC 1
#else
#define USE_ASYNC 0
#endif

typedef __attribute__((address_space(1))) int as1_int;  // global
typedef __attribute__((address_space(3))) int as3_int;  // LDS

__global__ __launch_bounds__(THREADS) void indrnn_stream(
    const float* __restrict__ x,     // [T, B, H]
    const float* __restrict__ w_hh,  // [H]
    const float* __restrict__ h0,    // [B, H]
    float* __restrict__ out)         // [T, B, H]
{
    const int n = blockIdx.x * THREADS + threadIdx.x;   // flat (b,h) lane
    const float w = w_hh[n & (HH - 1)];
    float h = h0[n];
    const float* xp = x + n;
    float*       op = out + n;

#if USE_ASYNC
    // 64 slots x 256 floats = 64 KB; lane i owns column i of every slot,
    // so each lane consumes exactly the word it async-loaded: per-wave
    // ASYNCcnt ordering is sufficient, no barriers, no cross-wave traffic.
    // Timestep u lives in ring slot (u & 63).
    __shared__ float ring[RING * THREADS];
    float* col = &ring[threadIdx.x];

    // Prologue: fill DEPTH slots.
#pragma unroll 1
    for (int t = 0; t < DEPTH; ++t) {
        __builtin_amdgcn_global_load_async_to_lds_b32(
            (as1_int*)(xp + (size_t)t * NN),
            (as3_int*)(col + t * THREADS),
            0, 0);
    }

    // Steady state: wait once for the UNROLL oldest copies (in-order
    // completion), consume 4 timesteps, refill 4. Outstanding returns to
    // DEPTH each iteration. Slot reuse distance = RING(64) > DEPTH(48).
#pragma unroll 1
    for (int t = 0; t < TT - DEPTH; t += UNROLL) {
        __builtin_amdgcn_s_wait_asynccnt(DEPTH - UNROLL);
#pragma unroll
        for (int i = 0; i < UNROLL; ++i) {
            float xv = col[((t + i) & (RING - 1)) * THREADS];
            h = fmaxf(fmaf(h, w, xv), 0.0f);
            __builtin_nontemporal_store(h, op + (size_t)(t + i) * NN);
        }
#pragma unroll
        for (int i = 0; i < UNROLL; ++i) {
            int u = t + DEPTH + i;
            __builtin_amdgcn_global_load_async_to_lds_b32(
                (as1_int*)(xp + (size_t)u * NN),
                (as3_int*)(col + (u & (RING - 1)) * THREADS),
                0, 0);
        }
    }

    // Tail: everything issued; drain without per-step waits.
    __builtin_amdgcn_s_wait_asynccnt(0);
#pragma unroll 4
    for (int t = TT - DEPTH; t < TT; ++t) {
        float xv = col[(t & (RING - 1)) * THREADS];
        h = fmaxf(fmaf(h, w, xv), 0.0f);
        __builtin_nontemporal_store(h, op + (size_t)t * NN);
    }
#else
    // Fallback: register-pipelined ping-pong streaming (8-deep prefetch).
    const int C = 8;
    float bufA[C], bufB[C];
#pragma unroll
    for (int i = 0; i < C; ++i)
        bufA[i] = xp[(size_t)i * NN];

#pragma unroll 1
    for (int t = 0; t < TT; t += 2 * C) {
#pragma unroll
        for (int i = 0; i < C; ++i)
            bufB[i] = xp[(size_t)(t + C + i) * NN];
#pragma unroll
        for (int i = 0; i < C; ++i) {
            h = fmaxf(fmaf(h, w, bufA[i]), 0.0f);
            __builtin_nontemporal_store(h, op + (size_t)(t + i) * NN);
        }
        if (t + 2 * C < TT) {
#pragma unroll
            for (int i = 0; i < C; ++i)
                bufA[i] = xp[(size_t)(t + 2 * C + i) * NN];
        }
#pragma unroll
        for (int i = 0; i < C; ++i) {
            h = fmaxf(fmaf(h, w, bufB[i]), 0.0f);
            __builtin_nontemporal_store(h, op + (size_t)(t + C + i) * NN);
        }
    }
#endif
}

extern "C" void kernel_launch(void* const* d_in, const int* in_sizes, int n_in,
                              void* d_out, int out_size, void* d_ws, size_t ws_size,
                              hipStream_t stream) {
    const float* x    = (const float*)d_in[0];  // [T,B,H]
    const float* w_hh = (const float*)d_in[1];  // [H]
    const float* h0   = (const float*)d_in[2];  // [B,H]
    float* out        = (float*)d_out;          // [T,B,H]

    dim3 grid(NN / THREADS);   // 256 blocks
    dim3 block(THREADS);       // 256 threads = 8 wave32
    indrnn_stream<<<grid, block, 0, stream>>>(x, w_hh, h0, out);
}